// SGWConv_58471684768128
// MI455X (gfx1250) — compile-verified
//
#include <hip/hip_runtime.h>

typedef __attribute__((ext_vector_type(2))) float v2f;
typedef __attribute__((ext_vector_type(8))) float v8f;

#define D 128  // D_IN == D_OUT == 128

// ---------------------------------------------------------------------------
// h = x @ W  via V_WMMA_F32_16X16X4_F32.
// Block = 256 threads = 8 waves; each wave computes one 16x16 tile of h.
// x is read exactly once -> non-temporal loads; W is hot (64 KB) -> regular.
// ---------------------------------------------------------------------------
__global__ void gemm_xw_wmma(const float* __restrict__ X,
                             const float* __restrict__ W,
                             float* __restrict__ H) {
  const int lane    = threadIdx.x & 31;
  const int wave    = threadIdx.x >> 5;       // 0..7  -> N tile
  const int rowBase = blockIdx.x * 16;
  const int nBase   = wave * 16;
  const int mn      = lane & 15;              // M index for A, N index for B/D
  const int kq      = (lane >> 4) * 2;        // 0 or 2 (K sub-offset per half-wave)

  // B fragments: b[kk] vgpr j <- W[kk*4 + kq + j][nBase + mn]
  v2f bfrag[32];
#pragma unroll
  for (int kk = 0; kk < 32; ++kk) {
    const int k = kk * 4 + kq;
    bfrag[kk].x = W[(k + 0) * D + nBase + mn];
    bfrag[kk].y = W[(k + 1) * D + nBase + mn];
  }

  v8f acc = {};
  const float* arow = X + (size_t)(rowBase + mn) * D + kq;
#pragma unroll
  for (int kk = 0; kk < 32; ++kk) {
    v2f a;
    a.x = __builtin_nontemporal_load(arow + kk * 4 + 0);
    a.y = __builtin_nontemporal_load(arow + kk * 4 + 1);
    // 8 args: (neg_a, A, neg_b, B, c_mod, C, reuse_a, reuse_b)
    acc = __builtin_amdgcn_wmma_f32_16x16x4_f32(
        false, a, false, bfrag[kk], (short)0, acc, false, false);
  }

  // D layout: VGPR v, lane l -> M = v + (l/16)*8, N = l%16
#pragma unroll
  for (int v = 0; v < 8; ++v) {
    const int row = rowBase + (lane >> 4) * 8 + v;
    H[(size_t)row * D + nBase + mn] = acc[v];
  }
}

// ---------------------------------------------------------------------------
// out[i][d] = bias[d]   (N*D threads)
// ---------------------------------------------------------------------------
__global__ void init_out(float* __restrict__ out, const float* __restrict__ bias,
                         int total) {
  int i = blockIdx.x * blockDim.x + threadIdx.x;
  if (i < total) out[i] = bias[i & (D - 1)];
}

__global__ void zero_buf(float* __restrict__ p, int total) {
  int i = blockIdx.x * blockDim.x + threadIdx.x;
  if (i < total) p[i] = 0.0f;
}

// ---------------------------------------------------------------------------
// Phase 1 SpMM:  y[row] += val * h[col]
// One wave32 per edge; lane handles a float4 slice (32*4 = 128 = D).
// Edge streams (rows/cols/vals) are single-use -> non-temporal loads, so the
// 192 MB L2 stays dedicated to the h/y/out working set (~78 MB resident).
// Scatter via native fp32 atomics (global_atomic_add_f32, L2-resident dest).
// ---------------------------------------------------------------------------
__global__ void spmm_scatter(const int* __restrict__ rows,
                             const int* __restrict__ cols,
                             const float* __restrict__ vals,
                             const float* __restrict__ H,
                             float* __restrict__ Y, int E) {
  const int e    = (int)((blockIdx.x * (unsigned)blockDim.x + threadIdx.x) >> 5);
  const int lane = threadIdx.x & 31;
  if (e >= E) return;
  const int   row = __builtin_nontemporal_load(rows + e);
  const int   col = __builtin_nontemporal_load(cols + e);
  const float v   = __builtin_nontemporal_load(vals + e);
  const float4 hv = ((const float4*)(H + (size_t)col * D))[lane];
  float* yp = Y + (size_t)row * D + lane * 4;
  unsafeAtomicAdd(yp + 0, v * hv.x);
  unsafeAtomicAdd(yp + 1, v * hv.y);
  unsafeAtomicAdd(yp + 2, v * hv.z);
  unsafeAtomicAdd(yp + 3, v * hv.w);
}

// ---------------------------------------------------------------------------
// Phase 2 SpMM with fused wavelet filter:
//   out[row] += val * filt_r[col] * y[col]
// ---------------------------------------------------------------------------
__global__ void spmm_scatter_filt(const int* __restrict__ rows,
                                  const int* __restrict__ cols,
                                  const float* __restrict__ vals,
                                  const float* __restrict__ Fr,   // filt + r*N
                                  const float* __restrict__ Y,
                                  float* __restrict__ out, int E) {
  const int e    = (int)((blockIdx.x * (unsigned)blockDim.x + threadIdx.x) >> 5);
  const int lane = threadIdx.x & 31;
  if (e >= E) return;
  const int   row = __builtin_nontemporal_load(rows + e);
  const int   col = __builtin_nontemporal_load(cols + e);
  const float v   = __builtin_nontemporal_load(vals + e);
  const float s   = v * Fr[col];
  const float4 yv = ((const float4*)(Y + (size_t)col * D))[lane];
  float* op = out + (size_t)row * D + lane * 4;
  unsafeAtomicAdd(op + 0, s * yv.x);
  unsafeAtomicAdd(op + 1, s * yv.y);
  unsafeAtomicAdd(op + 2, s * yv.z);
  unsafeAtomicAdd(op + 3, s * yv.w);
}

// ---------------------------------------------------------------------------
// Host-side orchestration.
// Inputs (setup_inputs order): x[N*128] f32, vals[R*E] f32, weight[128*128] f32,
// filt[R*N] f32, bias[128] f32, rows[R*E] i32, cols[R*E] i32.
// Workspace: h at d_ws (N*128 f32, 25.6 MB), y right after (25.6 MB).
// Scales processed sequentially so h + y + out (~77 MB) stay L2-resident.
// ---------------------------------------------------------------------------
extern "C" void kernel_launch(void* const* d_in, const int* in_sizes, int n_in,
                              void* d_out, int out_size, void* d_ws, size_t ws_size,
                              hipStream_t stream) {
  const float* x    = (const float*)d_in[0];
  const float* vals = (const float*)d_in[1];
  const float* W    = (const float*)d_in[2];
  const float* filt = (const float*)d_in[3];
  const float* bias = (const float*)d_in[4];
  const int*   rows = (const int*)d_in[5];
  const int*   cols = (const int*)d_in[6];
  float*       out  = (float*)d_out;

  const int N = in_sizes[0] / D;          // 50000
  const int R = 4;
  const int E = in_sizes[1] / R;          // 1600000
  const int ND = N * D;

  float* H = (float*)d_ws;
  float* Y = H + (size_t)ND;

  // 1) h = x @ W  (WMMA f32)
  gemm_xw_wmma<<<N / 16, 256, 0, stream>>>(x, W, H);

  // 2) out = bias (broadcast)
  init_out<<<(ND + 255) / 256, 256, 0, stream>>>(out, bias, ND);

  const int edgeBlocks = (E + 7) / 8;     // 8 edges (waves) per 256-thread block
  for (int r = 0; r < R; ++r) {
    zero_buf<<<(ND + 255) / 256, 256, 0, stream>>>(Y, ND);
    spmm_scatter<<<edgeBlocks, 256, 0, stream>>>(
        rows + (size_t)r * E, cols + (size_t)r * E, vals + (size_t)r * E, H, Y, E);
    spmm_scatter_filt<<<edgeBlocks, 256, 0, stream>>>(
        rows + (size_t)r * E, cols + (size_t)r * E, vals + (size_t)r * E,
        filt + (size_t)r * N, Y, out, E);
  }
}